// Noise_16226386444673
// MI455X (gfx1250) — compile-verified
//
#include <hip/hip_runtime.h>

typedef float v2f __attribute__((ext_vector_type(2)));
typedef float v8f __attribute__((ext_vector_type(8)));

// out[r,c,z] = sum_{i,j} T[(p,q),(i,j)] * in[r(i),c(j),z],  p/q = target bit of r/c.
// One V_WMMA_F32_16X16X4_F32 per 16 (d,z)-sites x 4 (p,q) outputs.
//   A (16x4): rows 0..3 = transfer matrix T (from kraus), rows 4..15 = 0
//   B (4x16): VGPR0 = {K0 lanes0-15, K2 lanes16-31} = {m00, m10}
//             VGPR1 = {K1 lanes0-15, K3 lanes16-31} = {m01, m11}
//   D rows 0..3 (lanes 0..15, vgprs 0..3) -> the four flip-combination stores.
__global__ __launch_bounds__(256) void kraus_channel_wmma(
    const float* __restrict__ in, const float* __restrict__ kraus,
    const int* __restrict__ tgt, const int* __restrict__ nqb,
    float* __restrict__ out,
    long long total, long long groups, int chunk, int nk)
{
    const int lane = threadIdx.x & 31;
    const int half = lane >> 4;      // 0: i=0 row block, 1: i=1 row block
    const int m    = lane & 15;      // A row / D column / site-in-group

    // ---- uniform geometry (all powers of two -> shifts) ----
    const int n   = nqb[0];
    const int t   = tgt[0];
    const int dim = 1 << n;
    const long long dd = (long long)dim * (long long)dim;
    const int B   = (int)(total / dd);           // batch (innermost)
    const int bsh = 31 - __clz(B);               // log2(B)
    const int rsh = n - t - 1;                   // log2(R), R = dim >> (t+1)
    const int gsh = rsh + bsh - 4;               // log2(groups per (r0,c_hi) region)
    const long long colFlip = 1LL << (rsh + bsh);            // flip target bit of col
    const long long rowFlip = ((long long)dim) << (rsh + bsh); // flip target bit of row

    // ---- A operand: T[(p,q),(i,j)] in rows 0..3 ----
    float a0 = 0.0f, a1 = 0.0f;
    if (m < 4) {
        const int p = m >> 1, q = m & 1;
        for (int kk = 0; kk < nk; ++kk) {
            const float* K = kraus + kk * 4;     // K[kk][2][2]
            const float kp = K[p * 2 + half];    // i = half
            a0 += kp * K[q * 2 + 0];             // j = 0  (K = 2*half)
            a1 += kp * K[q * 2 + 1];             // j = 1  (K = 2*half+1)
        }
    }
    v2f A; A.x = a0; A.y = a1;

    // ---- wave-uniform group range (force scalar address math) ----
    const int waveFlat = __builtin_amdgcn_readfirstlane(
        (int)((blockIdx.x * blockDim.x + threadIdx.x) >> 5));
    long long gid  = (long long)waveFlat * chunk;
    long long gend = gid + chunk;
    if (gend > groups) gend = groups;

    for (; gid < gend; ++gid) {
        const long long tg     = gid & ((1LL << gsh) - 1);   // 16-float group in region
        const long long region = gid >> gsh;
        const long long c_hi   = region & ((1LL << t) - 1);
        const long long ridx   = region >> t;
        const long long rlow   = ridx & ((1LL << rsh) - 1);
        const long long rhigh  = ridx >> rsh;
        const long long r0     = (rhigh << (rsh + 1)) + rlow;        // row with target bit 0
        const long long f00    = ((r0 * dim) + (c_hi << (rsh + 1))) << bsh;
        const long long inner  = (tg << 4);

        // B operand: one coalesced 64B segment per half-wave, twice.
        const long long base = f00 + inner + (half ? rowFlip : 0) + m;
        v2f Bm;
        Bm.x = __builtin_nontemporal_load(in + base);            // j=0 (m00 / m10)
        Bm.y = __builtin_nontemporal_load(in + base + colFlip);  // j=1 (m01 / m11)

        v8f C = {0.f, 0.f, 0.f, 0.f, 0.f, 0.f, 0.f, 0.f};
        v8f D = __builtin_amdgcn_wmma_f32_16x16x4_f32(
            /*neg_a=*/false, A, /*neg_b=*/false, Bm,
            /*c_mod=*/(short)0, C, /*reuse_a=*/false, /*reuse_b=*/false);

        if (half == 0) {                       // D rows 0..3 live in lanes 0..15
            const long long s = f00 + inner + m;
            __builtin_nontemporal_store(D[0], out + s);                       // (p,q)=(0,0)
            __builtin_nontemporal_store(D[1], out + s + colFlip);             // (0,1)
            __builtin_nontemporal_store(D[2], out + s + rowFlip);             // (1,0)
            __builtin_nontemporal_store(D[3], out + s + rowFlip + colFlip);   // (1,1)
        }
    }
}

extern "C" void kernel_launch(void* const* d_in, const int* in_sizes, int n_in,
                              void* d_out, int out_size, void* d_ws, size_t ws_size,
                              hipStream_t stream) {
    const float* state = (const float*)d_in[0];
    const float* kraus = (const float*)d_in[1];
    const int*   tgt   = (const int*)d_in[2];
    const int*   nqb   = (const int*)d_in[3];
    float*       out   = (float*)d_out;

    const long long total  = (long long)in_sizes[0];  // dim*dim*B
    const int       nk     = in_sizes[1] / 4;         // number of Kraus ops
    const long long groups = total / 64;              // 16 sites x 4 (p,q) per WMMA
    const int       chunk  = 16;                      // groups per wave (streams 1KB/1KB)
    const long long waves  = (groups + chunk - 1) / chunk;
    const long long thr    = waves * 32;
    const int       block  = 256;
    const long long blocks = (thr + block - 1) / block;

    kraus_channel_wmma<<<dim3((unsigned)blocks), dim3(block), 0, stream>>>(
        state, kraus, tgt, nqb, out, total, groups, chunk, nk);
}